// ResonanceCavity_85572928406137
// MI455X (gfx1250) — compile-verified
//
#include <hip/hip_runtime.h>
#include <math.h>
#include <stdint.h>

typedef float v2f __attribute__((ext_vector_type(2)));
typedef float v8f __attribute__((ext_vector_type(8)));

#define B_ROWS 8192
#define D_DIM  1024
#define K_RES  5
#define N1     1280   // K_RES * 256

// ---- CDNA5 async global->LDS copy (ASYNCcnt-tracked DMA), GVS addressing ----
__device__ __forceinline__ void async_b64(unsigned lds_off, const void* base, int voff) {
  asm volatile("global_load_async_to_lds_b64 %0, %1, %2"
               :: "v"(lds_off), "v"(voff), "s"(base) : "memory");
}
__device__ __forceinline__ void wait_async8() {
  asm volatile("s_wait_asynccnt 0x8" ::: "memory");
}
__device__ __forceinline__ void wait_async0() {
  asm volatile("s_wait_asynccnt 0x0" ::: "memory");
}

// ---------------- Stage 0: zero the gnorm_sq accumulator ----------------
__global__ void k_zero(float* __restrict__ p, int n) {
  int i = blockIdx.x * blockDim.x + threadIdx.x;
  if (i < n) p[i] = 0.0f;
}

// ---------------- Stage 1: A = h (8192x1024) x W^T (W is 1280x1024) ----------------
// 256 threads = 8 waves; tile 128(M)x128(N); K chunks of 16, double-buffered async LDS.
__global__ __launch_bounds__(256)
void k_gemm1(const float* __restrict__ h, const float* __restrict__ W,
             float* __restrict__ A) {
  __shared__ float hs[2][128][18];
  __shared__ float ws[2][128][18];
  const int tN = blockIdx.x * 128;
  const int tM = blockIdx.y * 128;
  const int t    = threadIdx.x;
  const int wave = t >> 5;
  const int lane = t & 31;
  const int half = lane >> 4;   // K half-select (ISA 16x4 f32 A layout)
  const int lr   = lane & 15;
  const int wrow = wave * 16;

  const v8f vz = {0.f,0.f,0.f,0.f,0.f,0.f,0.f,0.f};
  v8f acc[8];
  #pragma unroll
  for (int i = 0; i < 8; ++i) acc[i] = vz;

  const int f0 = t * 2;
  auto stage = [&](int buf, int k0) {
    #pragma unroll
    for (int i = 0; i < 4; ++i) {                 // 8 async b64 per thread/chunk
      const int f = i * 512 + f0;
      const int row = f >> 4, col = f & 15;       // 128x16 tile, col even -> 8B aligned
      async_b64((unsigned)(uintptr_t)&hs[buf][row][col], h,
                (((tM + row) * D_DIM) + k0 + col) * 4);
      async_b64((unsigned)(uintptr_t)&ws[buf][row][col], W,
                (((tN + row) * D_DIM) + k0 + col) * 4);
    }
  };

  stage(0, 0);
  for (int c = 0; c < 64; ++c) {
    if (c + 1 < 64) { stage((c + 1) & 1, (c + 1) * 16); wait_async8(); }
    else            { wait_async0(); }
    __syncthreads();
    const int buf = c & 1;
    #pragma unroll
    for (int kk = 0; kk < 4; ++kk) {
      v2f a;
      a.x = hs[buf][wrow + lr][kk*4 + half*2 + 0];
      a.y = hs[buf][wrow + lr][kk*4 + half*2 + 1];
      #pragma unroll
      for (int nt = 0; nt < 8; ++nt) {
        v2f b;
        b.x = ws[buf][nt*16 + lr][kk*4 + half*2 + 0];
        b.y = ws[buf][nt*16 + lr][kk*4 + half*2 + 1];
        acc[nt] = __builtin_amdgcn_wmma_f32_16x16x4_f32(
            false, a, false, b, (short)0, acc[nt], false, false);
      }
    }
    __syncthreads();
  }
  #pragma unroll
  for (int nt = 0; nt < 8; ++nt) {
    #pragma unroll
    for (int r = 0; r < 8; ++r) {
      const int row = tM + wrow + half*8 + r;
      const int col = tN + nt*16 + lr;
      A[(size_t)row * N1 + col] = acc[nt][r];
    }
  }
}

// ---------------- Stage 2: S = sum_k (A_k x M_k), plus per-(b,k) sum of squares ----
// 80 chunks (k,m0) streamed through a double-buffered async pipeline.
__global__ __launch_bounds__(256)
void k_gemm2(const float* __restrict__ A, const float* __restrict__ Mw,
             float* __restrict__ S, float* __restrict__ gsq) {
  __shared__ float as_[2][128][18];
  __shared__ float bs[2][16][128];
  const int tD = blockIdx.x * 128;
  const int tB = blockIdx.y * 128;
  const int t    = threadIdx.x;
  const int wave = t >> 5;
  const int lane = t & 31;
  const int half = lane >> 4;
  const int lr   = lane & 15;
  const int wrow = wave * 16;

  const v8f vz = {0.f,0.f,0.f,0.f,0.f,0.f,0.f,0.f};
  v8f ssum[8];
  #pragma unroll
  for (int i = 0; i < 8; ++i) ssum[i] = vz;
  v8f acc[8];

  const int f0 = t * 2;
  auto stage = [&](int buf, int c) {              // c = k*16 + m-chunk
    const int k  = c >> 4;
    const int m0 = (c & 15) * 16;
    #pragma unroll
    for (int i = 0; i < 4; ++i) {
      const int f = i * 512 + f0;
      { const int row = f >> 4, col = f & 15;     // A tile 128x16
        async_b64((unsigned)(uintptr_t)&as_[buf][row][col], A,
                  (((tB + row) * N1) + k*256 + m0 + col) * 4); }
      { const int row = f >> 7, col = f & 127;    // Mw tile 16x128, contiguous in LDS
        async_b64((unsigned)(uintptr_t)&bs[buf][row][col], Mw,
                  (((k*256 + m0 + row) * D_DIM) + tD + col) * 4); }
    }
  };

  stage(0, 0);
  for (int c = 0; c < 80; ++c) {
    const int k = c >> 4;
    if ((c & 15) == 0) {
      #pragma unroll
      for (int i = 0; i < 8; ++i) acc[i] = vz;
    }
    if (c + 1 < 80) { stage((c + 1) & 1, c + 1); wait_async8(); }
    else            { wait_async0(); }
    __syncthreads();
    const int buf = c & 1;
    #pragma unroll
    for (int kk = 0; kk < 4; ++kk) {
      v2f a;
      a.x = as_[buf][wrow + lr][kk*4 + half*2 + 0];
      a.y = as_[buf][wrow + lr][kk*4 + half*2 + 1];
      #pragma unroll
      for (int nt = 0; nt < 8; ++nt) {
        v2f b;
        b.x = bs[buf][kk*4 + half*2 + 0][nt*16 + lr];
        b.y = bs[buf][kk*4 + half*2 + 1][nt*16 + lr];
        acc[nt] = __builtin_amdgcn_wmma_f32_16x16x4_f32(
            false, a, false, b, (short)0, acc[nt], false, false);
      }
    }
    if ((c & 15) == 15) {
      // end of this k: fold into k-sum, reduce per-row sum of squares for the norm
      float sq[8];
      #pragma unroll
      for (int r = 0; r < 8; ++r) sq[r] = 0.f;
      #pragma unroll
      for (int nt = 0; nt < 8; ++nt) {
        ssum[nt] += acc[nt];
        #pragma unroll
        for (int r = 0; r < 8; ++r) sq[r] += acc[nt][r] * acc[nt][r];
      }
      #pragma unroll
      for (int r = 0; r < 8; ++r) {
        float v = sq[r];
        v += __shfl_xor(v, 1, 32);
        v += __shfl_xor(v, 2, 32);
        v += __shfl_xor(v, 4, 32);
        v += __shfl_xor(v, 8, 32);   // reduced within each 16-lane half
        if (lr == 0) {
          const int row = tB + wrow + half*8 + r;
          atomicAdd(&gsq[(size_t)row * K_RES + k], v);
        }
      }
    }
    __syncthreads();
  }
  #pragma unroll
  for (int nt = 0; nt < 8; ++nt) {
    #pragma unroll
    for (int r = 0; r < 8; ++r) {
      const int row = tB + wrow + half*8 + r;
      const int col = tD + nt*16 + lr;
      S[(size_t)row * D_DIM + col] = ssum[nt][r];
    }
  }
}

// ---------------- Stage 3: flux scale + breather + trust, fused output ------------
__global__ __launch_bounds__(256)
void k_final(const float* __restrict__ h, const float* __restrict__ residues,
             const float* __restrict__ gcve, const float* __restrict__ projw,
             const float* __restrict__ diffusivity, const float* __restrict__ trust_scores,
             const float* __restrict__ S, const float* __restrict__ gsq,
             float* __restrict__ out) {
  const int b = blockIdx.x;
  const int t = threadIdx.x;
  __shared__ float red[256];
  __shared__ float sh_xpos, sh_scale, sh_trust;

  const float4 hv = *(const float4*)(h + (size_t)b * D_DIM + t * 4);
  red[t] = hv.x + hv.y + hv.z + hv.w;
  __syncthreads();
  for (int off = 128; off > 0; off >>= 1) {
    if (t < off) red[t] += red[t + off];
    __syncthreads();
  }
  if (t == 0) {
    sh_xpos = red[0] * (1.0f / 1024.0f);
    const unsigned int HP[5] = {1000003u, 999983u, 1000033u, 1000037u, 1000039u};
    unsigned int us = 0u;
    #pragma unroll
    for (int j = 0; j < 5; ++j) {
      float q = residues[(size_t)b * 5 + j];
      q = fminf(fmaxf(q, -30000.0f), 30000.0f) * 100.0f;
      us += (unsigned int)(int)q * HP[j];
    }
    int idx = (int)us % 1280;
    if (idx < 0) idx += 1280;
    sh_trust = trust_scores[idx];
    float gmean = 0.f;
    #pragma unroll
    for (int k = 0; k < K_RES; ++k) gmean += sqrtf(gsq[(size_t)b * K_RES + k]);
    gmean *= 0.2f;
    const float flux = diffusivity[0] * gmean + 1e-6f;
    sh_scale = expf(-gcve[b] / flux);
  }
  __syncthreads();
  const float xpos  = sh_xpos;
  const float scale = sh_scale;
  const float trust = sh_trust;

  const float PRIMES[10] = {2.f,3.f,5.f,7.f,11.f,13.f,17.f,19.f,23.f,29.f};
  float amp[10];
  #pragma unroll
  for (int j = 0; j < 10; ++j) {
    const float om = 1.0f / PRIMES[j];
    const float sq = sqrtf(1.0f - om * om);
    const float c  = -3.14159265358979f + (float)j * (6.28318530717959f / 9.0f);
    const float sech = 1.0f / (coshf(sq * (xpos - c)) + 1e-8f);
    amp[j] = 4.0f * atanf((sq / (om + 1e-8f)) * sech * sinf(om * 0.1f));
  }

  const float4 sv = *(const float4*)(S + (size_t)b * D_DIM + t * 4);
  float ov[4] = {sv.x, sv.y, sv.z, sv.w};
  #pragma unroll
  for (int u = 0; u < 4; ++u) {
    const int d = t * 4 + u;
    float br = 0.0f;
    #pragma unroll
    for (int j = 0; j < 10; ++j) br += amp[j] * projw[(size_t)d * 10 + j];
    ov[u] = ov[u] * scale + 0.5f * br + 0.05f * trust;
  }
  float4 o; o.x = ov[0]; o.y = ov[1]; o.z = ov[2]; o.w = ov[3];
  *(float4*)(out + (size_t)b * D_DIM + t * 4) = o;
}

extern "C" void kernel_launch(void* const* d_in, const int* in_sizes, int n_in,
                              void* d_out, int out_size, void* d_ws, size_t ws_size,
                              hipStream_t stream) {
  const float* h          = (const float*)d_in[0];
  const float* residues   = (const float*)d_in[1];
  const float* gcve       = (const float*)d_in[2];
  const float* Mw         = (const float*)d_in[3];
  const float* coupling_w = (const float*)d_in[4];
  const float* projw      = (const float*)d_in[5];
  const float* diffus     = (const float*)d_in[6];
  const float* trust      = (const float*)d_in[7];
  float* out = (float*)d_out;

  float* A   = (float*)d_ws;                       // 8192*1280
  float* S   = A + (size_t)B_ROWS * N1;            // 8192*1024
  float* gsq = S + (size_t)B_ROWS * D_DIM;         // 8192*5

  const int ng = B_ROWS * K_RES;
  k_zero<<<(ng + 255) / 256, 256, 0, stream>>>(gsq, ng);
  k_gemm1<<<dim3(N1 / 128, B_ROWS / 128), 256, 0, stream>>>(h, coupling_w, A);
  k_gemm2<<<dim3(D_DIM / 128, B_ROWS / 128), 256, 0, stream>>>(A, Mw, S, gsq);
  k_final<<<B_ROWS, 256, 0, stream>>>(h, residues, gcve, projw, diffus, trust, S, gsq, out);
}